// TimeConvolutionReflectionPad_52158082843141
// MI455X (gfx1250) — compile-verified
//
#include <hip/hip_runtime.h>
#include <math.h>

// ---------------------------------------------------------------------------
// Four-step FFT (16384 = 128 x 128) where each 128-pt DFT stage is a complex
// 128x128x128 GEMM executed with V_WMMA_F32_16X16X4_F32 (full fp32 precision).
//
// Forward:  X[128d + c] = sum_b F[d][b] * ( w_N^{bc} * sum_a F[c][a] * A[a][b] )
//           with A[a][b] = x_pad[128a + b],  F[c][a] = exp(-2*pi*i*c*a/128)
// Inverse:  same structure with conjugated F / twiddles, 1/N scale, and only
//           output rows h < 64 (time index n = 128h + g < 8192).
//
// Layouts:
//  * F (DFT-128 matrix): row-major, stride 132  -> fragA pairs contiguous.
//  * Step-1 input A:     K-pair interleaved, (k,n) at (k>>1)*288 + 2n + (k&1)
//                        -> fragB pairs contiguous (single ds_load_b64),
//                        288 % 64 == 32 banks -> conflict-free across halves.
//  * Z (after twiddle):  row-major, stride 132, overlaid on A's buffer
//                        (single barrier separates all reads from writes);
//                        fragBT pairs contiguous.
// Register strategy: ~276 KB LDS pins us at 1 WG/WGP -> 2 waves/SIMD, so
// __launch_bounds__(256,1) lets the 8 tile-accumulator pairs (128 VGPRs) +
// hoisted fragments live entirely in registers (no scratch spills).
// ---------------------------------------------------------------------------

typedef float v2f __attribute__((ext_vector_type(2)));
typedef float v8f __attribute__((ext_vector_type(8)));

static constexpr int   kNF   = 16384;      // FFT length (2T)
static constexpr int   kT    = 8192;       // time length
static constexpr int   kLd   = 132;        // row stride for F and Z (floats)
static constexpr int   kLdP  = 288;        // pair-row stride for interleaved A
static constexpr int   kFSz  = 128 * kLd;  // 16896 floats
static constexpr int   kASz  = 64 * kLdP;  // 18432 floats (>= 128*132)
static constexpr float kTwoPiOver128 = 6.28318530717958647692f / 128.0f;
static constexpr float kTwoPiOverN   = 6.28318530717958647692f / 16384.0f;

__device__ __forceinline__ v8f wmma4(v2f a, v2f b, v8f c) {
  // D = A(16x4,f32) * B(4x16,f32) + C(16x16,f32)
  return __builtin_amdgcn_wmma_f32_16x16x4_f32(
      /*neg_a=*/false, a, /*neg_b=*/false, b,
      /*c_mod=*/(short)0, c, /*reuse_a=*/false, /*reuse_b=*/false);
}

// Interleaved address of element (k, n):  (k>>1)*kLdP + 2n + (k&1)
__device__ __forceinline__ int ailAddr(int k, int n) {
  return (k >> 1) * kLdP + 2 * n + (k & 1);
}

// A fragment: rows m0..m0+15, k-cols k0..k0+3 of row-major S (stride kLd).
// lane L: m = m0 + (L&15); VGPR j holds K = 2*(L>>4) + j  -> contiguous pair.
__device__ __forceinline__ v2f fragA(const float* __restrict__ S, int m0, int k0, int lane) {
  const float* p = S + (m0 + (lane & 15)) * kLd + (k0 + ((lane >> 4) << 1));
  v2f r; r.x = p[0]; r.y = p[1]; return r;
}

// B fragment from K-pair interleaved S: k-rows k0..k0+3, cols n0..n0+15.
// kb = k0 + 2*(L>>4) is even -> (kb,n),(kb+1,n) are adjacent: one b64 load.
__device__ __forceinline__ v2f fragBIL(const float* __restrict__ S, int k0, int n0, int lane) {
  int kb = k0 + ((lane >> 4) << 1);
  const float* p = S + (kb >> 1) * kLdP + 2 * (n0 + (lane & 15));
  v2f r; r.x = p[0]; r.y = p[1]; return r;
}

// B fragment of S^T (S row-major stride kLd): element (k, n) = S[n][k].
__device__ __forceinline__ v2f fragBT(const float* __restrict__ S, int k0, int n0, int lane) {
  const float* p = S + (n0 + (lane & 15)) * kLd + (k0 + ((lane >> 4) << 1));
  v2f r; r.x = p[0]; r.y = p[1]; return r;
}

// Build DFT-128 matrix F[c][a] = exp(-2*pi*i*c*a/128) into LDS.
__device__ __forceinline__ void buildF(float* Fr, float* Fi, int tid, int nthr) {
  for (int i = tid; i < 128 * 128; i += nthr) {
    int c = i >> 7, a = i & 127;
    float ang = (float)((c * a) & 127) * kTwoPiOver128;
    float s, co;
    __sincosf(ang, &s, &co);
    Fr[c * kLd + a] = co;
    Fi[c * kLd + a] = -s;
  }
}

// ---------------------------------------------------------------------------
// Step 1: inner 128-pt DFTs (over a) fused with long twiddle.
// Reads A in interleaved layout; after one barrier writes Z row-major over
// the same buffer. Wave `wid` owns row-tile c0 = wid*16 and accumulates all
// 8 column strips in registers (F fragment hoisted).
// ---------------------------------------------------------------------------
template <int SIGN>
__device__ __forceinline__ void innerFFTtwiddle(const float* __restrict__ Fr,
                                                const float* __restrict__ Fi,
                                                float* Ar, float* Ai,
                                                int wid, int lane) {
  const int c0 = wid * 16;
  v8f accR[8] = {};
  v8f accI[8] = {};
#pragma unroll 2
  for (int k0 = 0; k0 < 128; k0 += 4) {
    v2f far = fragA(Fr, c0, k0, lane);
    v2f fai = fragA(Fi, c0, k0, lane);
    v2f fs  = (SIGN < 0) ? -fai : fai;   // s * Fi
    v2f fn  = -fs;                       // -s * Fi
#pragma unroll
    for (int b = 0; b < 8; ++b) {
      v2f br = fragBIL(Ar, k0, b * 16, lane);
      v2f bi = fragBIL(Ai, k0, b * 16, lane);
      accR[b] = wmma4(far, br, accR[b]);
      accR[b] = wmma4(fn,  bi, accR[b]);
      accI[b] = wmma4(far, bi, accI[b]);
      accI[b] = wmma4(fs,  br, accI[b]);
    }
  }
  __syncthreads();                       // all interleaved reads of A done
  const int col = lane & 15, half = lane >> 4;
#pragma unroll
  for (int b = 0; b < 8; ++b) {
#pragma unroll
    for (int r = 0; r < 8; ++r) {
      int c  = c0 + r + 8 * half;
      int bb = b * 16 + col;
      float ang = (float)((bb * c) & (kNF - 1)) * kTwoPiOverN;
      float s, co;
      __sincosf(ang, &s, &co);
      float twi = (SIGN > 0) ? -s : s;   // w_N^{bc} fwd, w_N^{-bc} inv
      float yr = accR[b][r], yi = accI[b][r];
      Ar[c * kLd + bb] = yr * co - yi * twi;   // Z, row-major
      Ai[c * kLd + bb] = yr * twi + yi * co;
    }
  }
  __syncthreads();                       // Z fully written
}

// ---------------------------------------------------------------------------
// Forward: reflection-pad + FFT(16384) of sequence blockIdx.x -> xhat (ws)
// ---------------------------------------------------------------------------
__global__ void __launch_bounds__(256, 1)
fwd_fft_kernel(const float2* __restrict__ x,
               float2* __restrict__ xhat) {
  extern __shared__ float smem[];
  float* Fr = smem;
  float* Fi = Fr + kFSz;
  float* Ar = Fi + kFSz;
  float* Ai = Ar + kASz;

  const int tid  = threadIdx.x;
  const int nthr = blockDim.x;
  const int wid  = tid >> 5;
  const int lane = tid & 31;
  const int seq  = blockIdx.x;              // (c*Jr + jr)*A + a

  buildF(Fr, Fi, tid, nthr);

  // Load + reflection pad: x_pad[n] = n < T ? x[n] : x[2T-1-n];
  // A[a][b] = x_pad[128a + b], stored interleaved.
  const float2* xs = x + (size_t)seq * kT;
  for (int i = tid; i < kNF; i += nthr) {
    int t = (i < kT) ? i : (2 * kT - 1 - i);
    float2 v = xs[t];
    int ad = ailAddr(i >> 7, i & 127);
    Ar[ad] = v.x;
    Ai[ad] = v.y;
  }
  __syncthreads();

  innerFFTtwiddle<1>(Fr, Fi, Ar, Ai, wid, lane);

  // Step 2: Xmat[d][c] = (F * Z^T)[d][c] -> k = 128d + c (natural order).
  // Wave owns d0 = wid*16; F fragment hoisted across the 8 c-tiles.
  float2* dst = xhat + (size_t)seq * kNF;
  const int d0 = wid * 16;
  v8f accR[8] = {};
  v8f accI[8] = {};
#pragma unroll 2
  for (int k0 = 0; k0 < 128; k0 += 4) {
    v2f far = fragA(Fr, d0, k0, lane);
    v2f fai = fragA(Fi, d0, k0, lane);
    v2f fn  = -fai;
#pragma unroll
    for (int i = 0; i < 8; ++i) {
      v2f br = fragBT(Ar, k0, i * 16, lane);
      v2f bi = fragBT(Ai, k0, i * 16, lane);
      accR[i] = wmma4(far, br, accR[i]);
      accR[i] = wmma4(fn,  bi, accR[i]);
      accI[i] = wmma4(far, bi, accI[i]);
      accI[i] = wmma4(fai, br, accI[i]);
    }
  }
  const int col = lane & 15, half = lane >> 4;
#pragma unroll
  for (int i = 0; i < 8; ++i) {
#pragma unroll
    for (int r = 0; r < 8; ++r) {
      int d = d0 + r + 8 * half;
      int c = i * 16 + col;
      dst[d * 128 + c] = make_float2(accR[i][r], accI[i][r]);  // coalesced
    }
  }
}

// ---------------------------------------------------------------------------
// Inverse: gather spectrum + filter, pointwise multiply, IFFT, keep n < T.
// Output sequence o = (cch*CJ + j)*A + a
// ---------------------------------------------------------------------------
__global__ void __launch_bounds__(256, 1)
inv_fft_kernel(const float2* __restrict__ xhat,
               const float2* __restrict__ psi,
               const float2* __restrict__ phi,
               const long long* __restrict__ idx,
               float2* __restrict__ y) {
  extern __shared__ float smem[];
  float* Fr = smem;
  float* Fi = Fr + kFSz;
  float* Ar = Fi + kFSz;
  float* Ai = Ar + kASz;

  const int tid  = threadIdx.x;
  const int nthr = blockDim.x;
  const int wid  = tid >> 5;
  const int lane = tid & 31;

  const int o   = blockIdx.x;
  const int a   = o & 3;
  const int j   = (o >> 2) & 31;
  const int cch = o >> 7;
  const int jr  = (int)idx[2 * j];          // lambda gather (0..15)
  const int fi  = (int)idx[2 * j + 1];      // filter index into [psi;phi;phi]

  buildF(Fr, Fi, tid, nthr);

  const float2* xs = xhat + (size_t)(((cch * 16 + jr) * 4) + a) * kNF;
  const float2* fs = (fi < 24) ? (psi + (size_t)fi * kNF) : phi;

  // G[k] = xhat[k] * filt[k];  A[e][f] = G[128e + f], stored interleaved.
  for (int k = tid; k < kNF; k += nthr) {
    float2 xv = xs[k];
    float2 fv = fs[k];
    int ad = ailAddr(k >> 7, k & 127);
    Ar[ad] = xv.x * fv.x - xv.y * fv.y;
    Ai[ad] = xv.x * fv.y + xv.y * fv.x;
  }
  __syncthreads();

  innerFFTtwiddle<-1>(Fr, Fi, Ar, Ai, wid, lane);

  // Step 2 (conjugate): only h < 64 -> n = 128h + g < 8192.
  // Wave owns g-tile c0 = wid*16; the Z fragment is invariant across the 4
  // h-tiles, so hoist B and fold the conjugation into the B operands:
  //   Cr += Fr(x)Br + Fi(x)Bi ; Ci += Fr(x)Bi + Fi(x)(-Br)
  const float scale = 1.0f / 16384.0f;
  float2* dst = y + (size_t)o * kT;
  const int c0 = wid * 16;
  v8f accR[4] = {};
  v8f accI[4] = {};
#pragma unroll 2
  for (int k0 = 0; k0 < 128; k0 += 4) {
    v2f br  = fragBT(Ar, k0, c0, lane);
    v2f bi  = fragBT(Ai, k0, c0, lane);
    v2f brn = -br;
#pragma unroll
    for (int i = 0; i < 4; ++i) {
      v2f far = fragA(Fr, i * 16, k0, lane);
      v2f fai = fragA(Fi, i * 16, k0, lane);
      accR[i] = wmma4(far, br,  accR[i]);
      accR[i] = wmma4(fai, bi,  accR[i]);
      accI[i] = wmma4(far, bi,  accI[i]);
      accI[i] = wmma4(fai, brn, accI[i]);
    }
  }
  const int col = lane & 15, half = lane >> 4;
#pragma unroll
  for (int i = 0; i < 4; ++i) {
#pragma unroll
    for (int r = 0; r < 8; ++r) {
      int h = i * 16 + r + 8 * half;
      int g = c0 + col;
      dst[h * 128 + g] = make_float2(accR[i][r] * scale, accI[i][r] * scale);
    }
  }
}

extern "C" void kernel_launch(void* const* d_in, const int* in_sizes, int n_in,
                              void* d_out, int out_size, void* d_ws, size_t ws_size,
                              hipStream_t stream) {
  (void)in_sizes; (void)n_in; (void)out_size; (void)ws_size;
  const float2*    x   = (const float2*)d_in[0];     // [8,16,4,8192,2] f32
  const float2*    psi = (const float2*)d_in[1];     // [24,16384,2]    f32
  const float2*    phi = (const float2*)d_in[2];     // [1,16384,2]     f32
  const long long* idx = (const long long*)d_in[3];  // [32,2]          i64

  float2* xhat = (float2*)d_ws;                      // 512*16384*8 B = 64 MiB
  float2* y    = (float2*)d_out;                     // [8,32,4,8192,2] f32

  const size_t smemBytes = (size_t)(2 * kFSz + 2 * kASz) * sizeof(float); // ~276 KiB

  fwd_fft_kernel<<<dim3(512),  dim3(256), smemBytes, stream>>>(x, xhat);
  inv_fft_kernel<<<dim3(1024), dim3(256), smemBytes, stream>>>(xhat, psi, phi, idx, y);
}